// complex_MaxPool2D_32615981646097
// MI455X (gfx1250) — compile-verified
//
#include <hip/hip_runtime.h>
#include <stdint.h>

// Problem constants (from reference): x is [8,512,512,32] NHWC fp32, pool 2x2.
#define BB   8
#define HH   512
#define WW   512
#define CC   32
#define HO   256
#define WO   256
#define TW   32                 // output pixels per block tile
#define TIN  (2*TW)             // 64 input pixels per row segment
#define SEGF (TIN*CC)           // 2048 floats = 8KB per LDS segment
#define NSEG 4                  // re row0, re row1, im row0, im row1

typedef float f4 __attribute__((ext_vector_type(4)));

// CDNA5 async global->LDS copy (16B per lane). GVS addressing:
//   copies 16 bytes from global address saddr + vaddr to the LDS byte address in vdst
// Tracked with ASYNCcnt; completion enforced via s_wait_asynccnt.
__device__ __forceinline__ void async_ld_b128(uint32_t lds_byte,
                                              const float* gbase,
                                              uint32_t voff) {
  asm volatile("global_load_async_to_lds_b128 %0, %1, %2"
               :: "v"(lds_byte), "v"(voff), "s"((uint64_t)(uintptr_t)gbase)
               : "memory");
}

__device__ __forceinline__ void wait_async_all() {
  asm volatile("s_wait_asynccnt 0x0" ::: "memory");
}

__global__ __launch_bounds__(256)
void complex_maxpool2d_kernel(const float* __restrict__ xre,
                              const float* __restrict__ xim,
                              float* __restrict__ out) {
  __shared__ __align__(16) float smem[NSEG * SEGF];   // 32 KB

  const uint32_t tid = threadIdx.x;

  // Decompose block id: 8 tiles along Wo, 256 output rows, 8 batches.
  uint32_t t   = blockIdx.x;
  const uint32_t twx = t & (WO / TW - 1);  t >>= 3;    // tile x: 0..7
  const uint32_t ho  = t & (HO - 1);                   // output row
  const uint32_t b   = t >> 8;                         // batch

  const uint32_t h0  = ho * 2;
  const uint32_t wi0 = twx * TIN;                      // input col start
  const size_t row0 = (((size_t)b * HH + h0) * WW + wi0) * CC;
  const size_t row1 = row0 + (size_t)WW * CC;

  const float* seg_base[NSEG] = { xre + row0, xre + row1,
                                  xim + row0, xim + row1 };

  const uint32_t lds0 = (uint32_t)(uintptr_t)smem;     // LDS byte offset

  // Stage 32 KB: 2048 x b128 chunks; 8 async loads per thread, bases are
  // block-uniform SGPRs, consecutive lanes fetch consecutive 16B chunks.
#pragma unroll
  for (int s = 0; s < NSEG; ++s) {
#pragma unroll
    for (int k = 0; k < 2; ++k) {
      const uint32_t off16 = (uint32_t)(k * 256 + tid) * 16u;  // < 8192
      async_ld_b128(lds0 + (uint32_t)s * (SEGF * 4u) + off16,
                    seg_base[s], off16);
    }
  }

  wait_async_all();
  __syncthreads();

  // Each thread: one output pixel (p) x 4 channels (quad q).
  const uint32_t p   = tid >> 3;        // 0..31 output pixel in tile
  const uint32_t q   = tid & 7;         // channel quad
  const uint32_t ch  = q * 4;
  const uint32_t cx0 = 2 * p;
  const uint32_t cx1 = 2 * p + 1;

  const float* s_re0 = smem;
  const float* s_re1 = smem + SEGF;
  const float* s_im0 = smem + 2 * SEGF;
  const float* s_im1 = smem + 3 * SEGF;

  const f4 re00 = *(const f4*)(s_re0 + cx0 * CC + ch);
  const f4 re01 = *(const f4*)(s_re0 + cx1 * CC + ch);
  const f4 re10 = *(const f4*)(s_re1 + cx0 * CC + ch);
  const f4 re11 = *(const f4*)(s_re1 + cx1 * CC + ch);
  const f4 im00 = *(const f4*)(s_im0 + cx0 * CC + ch);
  const f4 im01 = *(const f4*)(s_im0 + cx1 * CC + ch);
  const f4 im10 = *(const f4*)(s_im1 + cx0 * CC + ch);
  const f4 im11 = *(const f4*)(s_im1 + cx1 * CC + ch);

  const f4 m00 = re00 * re00 + im00 * im00;
  const f4 m01 = re01 * re01 + im01 * im01;
  const f4 m10 = re10 * re10 + im10 * im10;
  const f4 m11 = re11 * re11 + im11 * im11;

  // Branchless first-max argmax in window order (0,0),(0,1),(1,0),(1,1).
  float ov[8];
#pragma unroll
  for (int j = 0; j < 4; ++j) {
    float m = m00[j], r = re00[j], i = im00[j];
    if (m01[j] > m) { m = m01[j]; r = re01[j]; i = im01[j]; }
    if (m10[j] > m) { m = m10[j]; r = re10[j]; i = im10[j]; }
    if (m11[j] > m) { m = m11[j]; r = re11[j]; i = im11[j]; }
    ov[2 * j]     = r;   // y_mag * cos(y_phase) == re at argmax
    ov[2 * j + 1] = i;   // y_mag * sin(y_phase) == im at argmax
  }

  const uint32_t wo = twx * TW + p;
  const size_t obase = ((((size_t)b * HO + ho) * WO + wo) * CC + ch) * 2;

  const f4 o0 = { ov[0], ov[1], ov[2], ov[3] };
  const f4 o1 = { ov[4], ov[5], ov[6], ov[7] };
  // Streaming stores: output never re-read, keep it out of near caches.
  __builtin_nontemporal_store(o0, (f4*)(out + obase));
  __builtin_nontemporal_store(o1, (f4*)(out + obase + 4));
}

extern "C" void kernel_launch(void* const* d_in, const int* in_sizes, int n_in,
                              void* d_out, int out_size, void* d_ws, size_t ws_size,
                              hipStream_t stream) {
  (void)in_sizes; (void)n_in; (void)out_size; (void)d_ws; (void)ws_size;
  const float* xre = (const float*)d_in[0];
  const float* xim = (const float*)d_in[1];
  float* out = (float*)d_out;

  const dim3 grid(BB * HO * (WO / TW));   // 8*256*8 = 16384 blocks
  const dim3 block(256);                  // 8 waves (wave32)
  complex_maxpool2d_kernel<<<grid, block, 0, stream>>>(xre, xim, out);
}